// PaddedDownsamplingTransform_49065706389615
// MI455X (gfx1250) — compile-verified
//
#include <hip/hip_runtime.h>
#include <hip/hip_bf16.h>
#include <stdint.h>

// Bicubic grid-sample, separable grid, reflect padding.
// B=16, C=8, H=W=512, fp32.
// One block per (b, y). Stage 4 reflected input rows x 8 channels (64KB)
// into LDS with gfx1250 async global->LDS loads, then 16-tap bicubic from LDS.

#define BQ 16
#define CQ 8
#define HQ 512
#define WQ 512
#define A_COEF (-0.75f)

__device__ __forceinline__ float cub_k1(float x) {
    // ((A+2)x - (A+3)) x^2 + 1, A=-0.75
    return (1.25f * x - 2.25f) * x * x + 1.0f;
}
__device__ __forceinline__ float cub_k2(float x) {
    // ((Ax - 5A)x + 8A)x - 4A, A=-0.75
    return ((-0.75f * x + 3.75f) * x - 6.0f) * x + 3.0f;
}
__device__ __forceinline__ int reflect511(int i) {
    // span = 511, period = 1022
    i = (i < 0) ? -i : i;
    i = i % 1022;
    if (i > 511) i = 1022 - i;
    return i;
}

extern "C" __global__ __launch_bounds__(256)
void bicubic_downsample_kernel(const float* __restrict__ x,
                               const float* __restrict__ rate,
                               const float* __restrict__ center,
                               float* __restrict__ out) {
    extern __shared__ float smem[];  // [4 rows][8 ch][512 x] = 64 KB

    const int tid = threadIdx.x;
    const int bid = blockIdx.x;          // b*512 + y
    const int b = bid >> 9;
    const int y = bid & 511;

    const float r  = rate[b];
    const float cx = center[2 * b + 0];
    const float cy = center[2 * b + 1];
    const float inv_r = 1.0f / r;

    // ---- vertical taps (uniform per block) ----
    const float gy  = (float)y * (1.0f / 256.0f) - 1.0f;
    const float iy  = ((gy - cy) * inv_r + cy + 1.0f) * 0.5f * 511.0f;
    const float yf  = floorf(iy);
    const float ty  = iy - yf;
    const int   y0  = (int)yf;
    float wy[4];
    wy[0] = cub_k2(ty + 1.0f);
    wy[1] = cub_k1(ty);
    wy[2] = cub_k1(1.0f - ty);
    wy[3] = cub_k2(2.0f - ty);
    int yi[4];
#pragma unroll
    for (int dy = 0; dy < 4; ++dy) yi[dy] = reflect511(y0 + dy - 1);

    // ---- async stage: 4 rows x 8 channels x 512 floats -> LDS ----
    // 4096 B128 chunks total; 16 per thread.
    const unsigned lds_base = (unsigned)(unsigned long long)(void*)smem;
#pragma unroll
    for (int i = 0; i < 16; ++i) {
        const int ch = tid + (i << 8);           // 0..4095
        const int rc = ch >> 7;                  // dy*8 + c  (0..31)
        const int dy = rc >> 3;
        const int c  = rc & 7;
        const int xq = (ch & 127) << 2;          // dword index within row
        const unsigned gofs =
            (unsigned)(((((b * CQ + c) * HQ + yi[dy]) * WQ) + xq) * 4);
        const unsigned lofs = lds_base + ((unsigned)ch << 4);
        asm volatile("global_load_async_to_lds_b128 %0, %1, %2 offset:0"
                     :
                     : "v"(lofs), "v"(gofs), "s"(x)
                     : "memory");
    }
    asm volatile("s_wait_asynccnt 0x0" ::: "memory");
    __syncthreads();

    // ---- compute: 2 output x positions per thread, 8 channels each ----
#pragma unroll
    for (int xx = 0; xx < 2; ++xx) {
        const int xo = tid + (xx << 8);
        const float gx  = (float)xo * (1.0f / 256.0f) - 1.0f;
        const float ix  = ((gx - cx) * inv_r + cx + 1.0f) * 0.5f * 511.0f;
        const float xf  = floorf(ix);
        const float tx  = ix - xf;
        const int   x0  = (int)xf;
        const float wx0 = cub_k2(tx + 1.0f);
        const float wx1 = cub_k1(tx);
        const float wx2 = cub_k1(1.0f - tx);
        const float wx3 = cub_k2(2.0f - tx);
        const int xi0 = reflect511(x0 - 1);
        const int xi1 = reflect511(x0);
        const int xi2 = reflect511(x0 + 1);
        const int xi3 = reflect511(x0 + 2);

        const size_t obase = (((size_t)(b * CQ) * HQ + (size_t)y) * WQ) + (size_t)xo;
#pragma unroll
        for (int c = 0; c < 8; ++c) {
            const float* rp = smem + c * WQ;     // + dy*(8*512)
            float acc = 0.0f;
#pragma unroll
            for (int dy = 0; dy < 4; ++dy) {
                const float* q = rp + dy * (CQ * WQ);
                float s = wx0 * q[xi0];
                s = fmaf(wx1, q[xi1], s);
                s = fmaf(wx2, q[xi2], s);
                s = fmaf(wx3, q[xi3], s);
                acc = fmaf(wy[dy], s, acc);
            }
            out[obase + (size_t)c * (HQ * WQ)] = acc;
        }
    }
}

extern "C" void kernel_launch(void* const* d_in, const int* in_sizes, int n_in,
                              void* d_out, int out_size, void* d_ws, size_t ws_size,
                              hipStream_t stream) {
    (void)in_sizes; (void)n_in; (void)out_size; (void)d_ws; (void)ws_size;
    const float* x      = (const float*)d_in[0];
    const float* rate   = (const float*)d_in[1];
    const float* center = (const float*)d_in[2];
    float* out = (float*)d_out;

    const dim3 grid(BQ * HQ);       // 8192 blocks: one per (b, y)
    const dim3 block(256);
    const size_t shmem = 4 * CQ * WQ * sizeof(float);  // 64 KB
    bicubic_downsample_kernel<<<grid, block, shmem, stream>>>(x, rate, center, out);
}